// RelativePositionEncoding_85478439125479
// MI455X (gfx1250) — compile-verified
//
#include <hip/hip_runtime.h>
#include <cstdint>

#define NTOK 1024
#define CZ   128
#define RMAX 32
#define SMAX 2
#define NRES (2 * RMAX + 2)              // 66 residue classes
#define NCHN (2 * SMAX + 2)              // 6 chain classes
#define TOTD (2 * NRES + 1 + NCHN)       // 139 table rows
#define WT_FLOATS (TOTD * CZ)            // 17792
#define WT_BYTES  (WT_FLOATS * 4)        // 71168 (16B-aligned: /16 = 4448)
#define PK_BYTES  (NTOK * 16)            // 16384 packed descriptors
#define SMEM_BYTES (WT_BYTES + PK_BYTES) // 87552
#define SMEM_CHUNKS (SMEM_BYTES / 16)    // 5472 b128 transfers

typedef __attribute__((ext_vector_type(4))) float v4f;
typedef __attribute__((ext_vector_type(4))) int   v4i;

__device__ __forceinline__ int iclamp(int x, int lo, int hi) {
  return x < lo ? lo : (x > hi ? hi : x);
}

// Prep kernel: build the contiguous LDS image in d_ws:
//   [ Wt: transposed weights, TOTD x CZ row-major (71168 B) |
//     pk: per-token descriptor int4 {tok, res, asym, ent | sym<<16} (16384 B) ]
// One-time cost; makes the main kernel's async fill and all gathers coalesced.
__global__ __launch_bounds__(256) void prep(const float* __restrict__ W,
                                            const int* __restrict__ tok,
                                            const int* __restrict__ res,
                                            const int* __restrict__ asym,
                                            const int* __restrict__ ent,
                                            const int* __restrict__ sym,
                                            float* __restrict__ ws) {
  int t = blockIdx.x * 256 + threadIdx.x;
  if (t < WT_FLOATS) {
    int d = t / CZ;
    int c = t - d * CZ;
    ws[t] = W[c * TOTD + d];
  } else if (t < WT_FLOATS + NTOK) {
    int j = t - WT_FLOATS;
    v4i p;
    p.x = tok[j];
    p.y = res[j];
    p.z = asym[j];
    p.w = (ent[j] & 0xFFFF) | (sym[j] << 16);
    *(v4i*)(ws + WT_FLOATS + (size_t)j * 4) = p;
  }
}

// Main kernel: one block per output row i; 8 waves x 32 lanes (wave32).
// Wave w handles j in [w*128, w*128+128); lane l owns channels 4l..4l+3.
// Steady-state loop touches only LDS + one non-temporal b128 store.
__global__ __launch_bounds__(256) void relpos(const float* __restrict__ ws,
                                              float* __restrict__ out) {
  __shared__ __align__(16) unsigned char smem[SMEM_BYTES]; // 87552 B of 320 KB/WGP
  float*     sWt = (float*)smem;
  const v4i* sPk = (const v4i*)(smem + WT_BYTES);

  const int tid = threadIdx.x;
  const int i   = blockIdx.x;

  // --- CDNA5 async global->LDS fill (ASYNCcnt path, bypasses VGPRs) ---
  // GVS mode: mem_addr = SGPR_base + VGPR_offset; LDS byte addr in VDST VGPR.
  {
    const uint32_t lds0  = (uint32_t)(uintptr_t)(&smem[0]);
    const uint64_t gbase = (uint64_t)(uintptr_t)ws;
    for (int k = tid; k < SMEM_CHUNKS; k += 256) {
      const uint32_t boff = (uint32_t)k * 16u;
      const uint32_t ldsa = lds0 + boff;
      asm volatile("global_load_async_to_lds_b128 %0, %1, %2"
                   :: "v"(ldsa), "v"(boff), "s"(gbase)
                   : "memory");
    }
    asm volatile("s_wait_asynccnt 0" ::: "memory");
  }
  __syncthreads();

  const int wave = tid >> 5;
  const int lane = tid & 31;
  const int c0   = lane << 2;          // channel base: 0..124, step 4

  // Row-i scalars from LDS (broadcast read).
  const v4i pi    = sPk[i];
  const int tok_i = pi.x;
  const int res_i = pi.y;
  const int asym_i = pi.z;
  const int ent_i = pi.w & 0xFFFF;
  const int sym_i = pi.w >> 16;

  // Entity row (132) uses the same channels every iteration -> hoist.
  const v4f went = *(const v4f*)&sWt[2 * NRES * CZ + c0];

  float* orow = out + (size_t)i * NTOK * CZ;

  const int jbase = wave * (NTOK / 8);
#pragma unroll 4
  for (int jj = 0; jj < NTOK / 8; ++jj) {
    const int j = jbase + jj;

    // One broadcast b128 LDS read fetches all per-j metadata.
    const v4i p  = sPk[j];
    const int tj = p.x;
    const int rj = p.y;
    const int aj = p.z;
    const int ej = p.w & 0xFFFF;
    const int sj = p.w >> 16;

    const bool  sc = (asym_i == aj);
    const bool  sr = (res_i == rj);
    const float ef = (ent_i == ej) ? 1.0f : 0.0f;

    const int dres = sc         ? iclamp(res_i - rj + RMAX, 0, 2 * RMAX) : 2 * RMAX + 1;
    const int dtok = (sc && sr) ? iclamp(tok_i - tj + RMAX, 0, 2 * RMAX) : 2 * RMAX + 1;
    const int dchn = sc         ? (2 * SMAX + 1)
                                : iclamp(sym_i - sj + SMAX, 0, 2 * SMAX);

    // Three dynamic row gathers (b128, conflict-free: consecutive lanes read
    // consecutive 16B chunks of a 512B row).
    const v4f a  = *(const v4f*)&sWt[dres * CZ + c0];
    const v4f b  = *(const v4f*)&sWt[(NRES + dtok) * CZ + c0];
    const v4f ch = *(const v4f*)&sWt[(2 * NRES + 1 + dchn) * CZ + c0];

    v4f r = a + b + ch + ef * went;

    // Output is write-once streaming: non-temporal b128 store.
    __builtin_nontemporal_store(r, (v4f*)&orow[(size_t)j * CZ + c0]);
  }
}

extern "C" void kernel_launch(void* const* d_in, const int* in_sizes, int n_in,
                              void* d_out, int out_size, void* d_ws, size_t ws_size,
                              hipStream_t stream) {
  const int*   tok  = (const int*)d_in[0];
  const int*   res  = (const int*)d_in[1];
  const int*   asym = (const int*)d_in[2];
  const int*   ent  = (const int*)d_in[3];
  const int*   sym  = (const int*)d_in[4];
  const float* W    = (const float*)d_in[5];
  float*       out  = (float*)d_out;
  float*       ws   = (float*)d_ws;    // needs 87552 B of scratch

  const int nPrep = (WT_FLOATS + NTOK + 255) / 256;  // 74 blocks
  hipLaunchKernelGGL(prep,   dim3(nPrep), dim3(256), 0, stream,
                     W, tok, res, asym, ent, sym, ws);
  hipLaunchKernelGGL(relpos, dim3(NTOK),  dim3(256), 0, stream, ws, out);
}